// MINDLoss3D_83769042141660
// MI455X (gfx1250) — compile-verified
//
#include <hip/hip_runtime.h>
#include <float.h>

#define VOL 96
#define TZ 8
#define TY 16
#define TX 16
#define EZ (TZ + 4)
#define EY (TY + 4)
#define EX (TX + 4)
#define NTHREADS 256
#define NCH 12
#define NGROUPS ((EZ * EY) / 16)   // 240 rows / 16 = 15 WMMA row-groups

typedef float v2f __attribute__((ext_vector_type(2)));
typedef float v8f __attribute__((ext_vector_type(8)));

// Shift offsets (z,y,x for A and B of each pair), = (idx-1)*dilation, dilation=2.
// Derived from the reference's six-point neighbourhood and dist==2 pair mask.
__constant__ int OFFS[NCH][6] = {
  { 0, 0,-2,  -2, 0, 0},
  { 0,-2, 0,  -2, 0, 0},
  { 0,-2, 0,   0, 0,-2},
  { 0, 0, 2,  -2, 0, 0},
  { 0, 0, 2,   0,-2, 0},
  { 2, 0, 0,   0, 0,-2},
  { 2, 0, 0,   0,-2, 0},
  { 2, 0, 0,   0, 0, 2},
  { 0, 2, 0,  -2, 0, 0},
  { 0, 2, 0,   0, 0,-2},
  { 0, 2, 0,   0, 0, 2},
  { 0, 2, 0,   2, 0, 0},
};

struct SmemT {
  float sq[EZ][EY][EX];   // squared diffs, halo-extended     (19.2 KB)
  float xs[EZ][EY][TX];   // after x box-sum (WMMA)           (15.4 KB)
  float ys[EZ][TY][TX];   // after y box-sum                  (12.3 KB)
  float red[NTHREADS];    // block reduction                  ( 1.0 KB)
};

__device__ __forceinline__ int clampi(int v, int lo, int hi) {
  return v < lo ? lo : (v > hi ? hi : v);
}

// Banded 0/1 matrix S(20x16): S[k][n] = 1 iff n <= k <= n+4.
// B-matrix VGPR layout for f32 4x16 (analogous to C/D: VGPR v holds rows v, v+2):
// lane -> N = lane&15, VGPR0 -> K = 2*(lane>>4)+0 ... chunk j adds 4*j.
__device__ __forceinline__ void make_bmat(v2f* bmat) {
  const int lane = threadIdx.x & 31;
  const int n = lane & 15;
  const int ks = (lane >> 4) << 1;
#pragma unroll
  for (int j = 0; j < 5; ++j) {
    const int k0 = 4 * j + ks;
    const int k1 = k0 + 1;
    bmat[j].x = (n <= k0 && k0 <= n + 4) ? 1.0f : 0.0f;
    bmat[j].y = (n <= k1 && k1 <= n + 4) ? 1.0f : 0.0f;
  }
}

// Compute SSD (box-mean of squared shifted diff) for channel c over a
// TZxTYxTX tile; result per-thread in ssd[TZ] (thread owns y=t>>4, x=t&15).
__device__ void ssd_channel(const float* __restrict__ img,
                            int z0, int y0, int x0, int c,
                            SmemT* L, const v2f* bmat, float* ssd) {
  const int t    = threadIdx.x;
  const int lane = t & 31;
  const int wave = t >> 5;
  __syncthreads();  // protect previous channel's LDS consumers

  const int az = OFFS[c][0], ay = OFFS[c][1], ax = OFFS[c][2];
  const int bz = OFFS[c][3], by = OFFS[c][4], bx = OFFS[c][5];
  float* sqf = &L->sq[0][0][0];
  for (int e = t; e < EZ * EY * EX; e += NTHREADS) {
    const int lz = e / (EY * EX);
    const int r  = e - lz * (EY * EX);
    const int ly = r / EX;
    const int lx = r - ly * EX;
    // edge-clamp twice: box-pad clamp, then shift clamp (matches reference)
    const int gz = clampi(z0 - 2 + lz, 0, VOL - 1);
    const int gy = clampi(y0 - 2 + ly, 0, VOL - 1);
    const int gx = clampi(x0 - 2 + lx, 0, VOL - 1);
    const int pz = clampi(gz + az, 0, VOL - 1);
    const int py = clampi(gy + ay, 0, VOL - 1);
    const int px = clampi(gx + ax, 0, VOL - 1);
    const int qz = clampi(gz + bz, 0, VOL - 1);
    const int qy = clampi(gy + by, 0, VOL - 1);
    const int qx = clampi(gx + bx, 0, VOL - 1);
    const float va = img[(pz * VOL + py) * VOL + px];
    const float vb = img[(qz * VOL + qy) * VOL + qx];
    const float d  = va - vb;
    sqf[e] = d * d;
  }
  __syncthreads();

  // x-direction 5-tap box sum via V_WMMA_F32_16X16X4_F32:
  // rows = flattened (z,y) pairs (240 -> 15 groups of 16), cols = x.
  {
    float* xsf = &L->xs[0][0][0];
    const int n     = lane & 15;          // M for A, N for B/D
    const int ks    = (lane >> 4) << 1;   // K selector (0 or 2)
    const int mbase = (lane >> 4) << 3;   // D rows i / i+8
    for (int g = wave; g < NGROUPS; g += (NTHREADS / 32)) {
      const int row = g * 16 + n;
      v8f acc = {0.f, 0.f, 0.f, 0.f, 0.f, 0.f, 0.f, 0.f};
#pragma unroll
      for (int j = 0; j < 5; ++j) {
        v2f a;
        a.x = sqf[row * EX + 4 * j + ks + 0];
        a.y = sqf[row * EX + 4 * j + ks + 1];
        acc = __builtin_amdgcn_wmma_f32_16x16x4_f32(
            false, a, false, bmat[j], (short)0, acc, false, false);
      }
#pragma unroll
      for (int i = 0; i < 8; ++i) {
        const int R = g * 16 + mbase + i;
        xsf[R * TX + n] = acc[i];
      }
    }
  }
  __syncthreads();

  // y-direction 5-tap box sum
  for (int e = t; e < EZ * TY * TX; e += NTHREADS) {
    const int lz = e / (TY * TX);
    const int r  = e - lz * (TY * TX);
    const int ly = r / TX;
    const int lx = r - ly * TX;
    float s = 0.f;
#pragma unroll
    for (int j = 0; j < 5; ++j) s += L->xs[lz][ly + j][lx];
    L->ys[lz][ly][lx] = s;
  }
  __syncthreads();

  // z-direction 5-tap box sum -> registers (thread owns y=t>>4, x=t&15)
  const int yy = t >> 4;
  const int xx = t & 15;
#pragma unroll
  for (int z = 0; z < TZ; ++z) {
    float s = 0.f;
#pragma unroll
    for (int j = 0; j < 5; ++j) s += L->ys[z + j][yy][xx];
    ssd[z] = s * (1.0f / 125.0f);
  }
}

__device__ __forceinline__ void tile_origin(int* x0, int* y0, int* z0) {
  const int tile = blockIdx.x;
  *x0 = (tile % 6) * TX;
  *y0 = ((tile / 6) % 6) * TY;
  *z0 = (tile / 36) * TZ;
}

__device__ __forceinline__ float block_reduce(SmemT* L, float v) {
  const int t = threadIdx.x;
  __syncthreads();
  L->red[t] = v;
  __syncthreads();
  for (int s = NTHREADS / 2; s > 0; s >>= 1) {
    if (t < s) L->red[t] += L->red[t + s];
    __syncthreads();
  }
  return L->red[0];
}

// Pass A: accumulate sum of mind_var per image (for gm)
__global__ __launch_bounds__(NTHREADS)
void mind_var_kernel(const float* __restrict__ in0,
                     const float* __restrict__ in1,
                     float* __restrict__ varsum) {
  __shared__ SmemT L;
  int x0, y0, z0;
  tile_origin(&x0, &y0, &z0);
  const int b  = blockIdx.y;
  const int im = blockIdx.z;
  const float* img = ((im == 0) ? in0 : in1) + (size_t)b * VOL * VOL * VOL;

  v2f bmat[5];
  make_bmat(bmat);

  float mn[TZ], sm[TZ], ssd[TZ];
#pragma unroll
  for (int z = 0; z < TZ; ++z) { mn[z] = FLT_MAX; sm[z] = 0.f; }
  for (int c = 0; c < NCH; ++c) {
    ssd_channel(img, z0, y0, x0, c, &L, bmat, ssd);
#pragma unroll
    for (int z = 0; z < TZ; ++z) {
      mn[z] = fminf(mn[z], ssd[z]);
      sm[z] += ssd[z];
    }
  }
  float part = 0.f;
#pragma unroll
  for (int z = 0; z < TZ; ++z)
    part += sm[z] * (1.0f / 12.0f) - mn[z];   // mean_c(ssd - min) = mean - min

  const float total = block_reduce(&L, part);
  if (threadIdx.x == 0) atomicAdd(&varsum[im], total);
}

// Pass B: recompute SSD, apply clip/exp, accumulate MSE
__global__ __launch_bounds__(NTHREADS)
void mind_mse_kernel(const float* __restrict__ in0,
                     const float* __restrict__ in1,
                     const float* __restrict__ varsum,
                     float* __restrict__ out) {
  __shared__ SmemT L;
  int x0, y0, z0;
  tile_origin(&x0, &y0, &z0);
  const int b = blockIdx.y;
  const float* imgs[2] = { in0 + (size_t)b * VOL * VOL * VOL,
                           in1 + (size_t)b * VOL * VOL * VOL };
  const float kNvar = 1.0f / (2.0f * VOL * VOL * VOL);  // 1/1769472
  const float gm[2] = { varsum[0] * kNvar, varsum[1] * kNvar };

  v2f bmat[5];
  make_bmat(bmat);

  float mn[2][TZ], vr[2][TZ], ssd[TZ];
  for (int im = 0; im < 2; ++im) {
    float sm[TZ];
#pragma unroll
    for (int z = 0; z < TZ; ++z) { mn[im][z] = FLT_MAX; sm[z] = 0.f; }
    for (int c = 0; c < NCH; ++c) {
      ssd_channel(imgs[im], z0, y0, x0, c, &L, bmat, ssd);
#pragma unroll
      for (int z = 0; z < TZ; ++z) {
        mn[im][z] = fminf(mn[im][z], ssd[z]);
        sm[z] += ssd[z];
      }
    }
#pragma unroll
    for (int z = 0; z < TZ; ++z) {
      const float v = sm[z] * (1.0f / 12.0f) - mn[im][z];
      vr[im][z] = fminf(fmaxf(v, gm[im] * 0.001f), gm[im] * 1000.0f);
    }
  }

  float acc = 0.f;
  float s0[TZ], s1[TZ];
  for (int c = 0; c < NCH; ++c) {
    ssd_channel(imgs[0], z0, y0, x0, c, &L, bmat, s0);
    ssd_channel(imgs[1], z0, y0, x0, c, &L, bmat, s1);
#pragma unroll
    for (int z = 0; z < TZ; ++z) {
      const float e0 = __expf(-(s0[z] - mn[0][z]) / vr[0][z]);
      const float e1 = __expf(-(s1[z] - mn[1][z]) / vr[1][z]);
      const float d  = e0 - e1;
      acc += d * d;
    }
  }

  const float total = block_reduce(&L, acc);
  const float kNmse = 1.0f / (2.0f * NCH * VOL * VOL * VOL);  // 1/21233664
  if (threadIdx.x == 0) atomicAdd(out, total * kNmse);
}

extern "C" void kernel_launch(void* const* d_in, const int* in_sizes, int n_in,
                              void* d_out, int out_size, void* d_ws, size_t ws_size,
                              hipStream_t stream) {
  const float* in0 = (const float*)d_in[0];   // input  [2,1,96,96,96] f32
  const float* in1 = (const float*)d_in[1];   // target [2,1,96,96,96] f32
  float* out = (float*)d_out;                 // scalar f32
  float* ws  = (float*)d_ws;                  // ws[0..1] = per-image var sums

  hipMemsetAsync(ws, 0, 2 * sizeof(float), stream);
  hipMemsetAsync(out, 0, sizeof(float), stream);

  const dim3 block(NTHREADS);
  const dim3 gridA(6 * 6 * 12, 2, 2);  // tiles x batch x image
  const dim3 gridB(6 * 6 * 12, 2, 1);  // tiles x batch
  mind_var_kernel<<<gridA, block, 0, stream>>>(in0, in1, ws);
  mind_mse_kernel<<<gridB, block, 0, stream>>>(in0, in1, ws, out);
}